// SwitchRouter_6674379178473
// MI455X (gfx1250) — compile-verified
//
#include <hip/hip_runtime.h>
#include <math.h>

#define NTOK 32768          // 8 * 4096 tokens
#define CDIM 896
#define NEXP 3
#define CAP  13653          // int(1.25 * 32768 / 3)
#define TOK_PER_BLK 64
#define NBLK (NTOK / TOK_PER_BLK)   // 512
#define KCHUNK 64
#define NCHUNK (CDIM / KCHUNK)      // 14
#define XS_STRIDE 68                // conflict-free ds_load_b64 of A fragments
#define XS_SIZE (64 * XS_STRIDE)
#define W_STRIDE 900                // expert rows on distinct LDS banks

typedef float v2f __attribute__((ext_vector_type(2)));
typedef float v8f __attribute__((ext_vector_type(8)));

#if defined(__gfx1250__) && __has_builtin(__builtin_amdgcn_wmma_f32_16x16x4_f32)
#define USE_WMMA 1
#else
#define USE_WMMA 0
#endif

#if defined(__gfx1250__)
#define GFX_ASYNC 1
typedef int v4i_vs __attribute__((vector_size(4 * sizeof(int))));
typedef __attribute__((address_space(1))) v4i_vs as1_v4i;
typedef __attribute__((address_space(3))) v4i_vs as3_v4i;
#if __has_builtin(__builtin_amdgcn_s_wait_asynccnt)
#define WAIT_ASYNC(n) __builtin_amdgcn_s_wait_asynccnt(n)
#else
#define WAIT_ASYNC(n) asm volatile("s_wait_asynccnt %0" ::"i"(n) : "memory")
#endif
#else
#define GFX_ASYNC 0
#define WAIT_ASYNC(n) ((void)0)
#endif

#if GFX_ASYNC
// One 16-byte memory->LDS async copy (per-lane addresses), ASYNCcnt-tracked.
__device__ __forceinline__ void async_copy_b128(const float* g, float* l)
{
#if __has_builtin(__builtin_amdgcn_global_load_async_to_lds_b128)
    __builtin_amdgcn_global_load_async_to_lds_b128(
        (as1_v4i*)g, (as3_v4i*)l, 0, 0);
#else
    unsigned loff = (unsigned)(unsigned long long)(as3_v4i*)l;  // LDS byte offset
    asm volatile("global_load_async_to_lds_b128 %0, %1, off"
                 ::"v"(loff), "v"(g) : "memory");
#endif
}
#endif

// ---------------------------------------------------------------------------
// Kernel 1: router GEMM (WMMA f32 16x16x4) + softmax/argmax + block partials.
// Block = 128 threads (4 wave32), 64 tokens/block, double-buffered async tiles.
// ---------------------------------------------------------------------------
__global__ void __launch_bounds__(128)
router_kernel(const float* __restrict__ x, const float* __restrict__ W,
              float* __restrict__ probs, int* __restrict__ eidx,
              float* __restrict__ pz, float* __restrict__ pP, int* __restrict__ pc)
{
    __shared__ float xs[2 * XS_SIZE];        // double-buffered 64x64 x-tile
    __shared__ float lw[NEXP * W_STRIDE];    // full 3x896 router matrix
    __shared__ float lg[64][3];              // logits per token
    __shared__ float redf[4][64];            // z, p0, p1, p2 partials
    __shared__ int   redi[3][64];            // per-expert assignment counts

    const int tid = threadIdx.x;
    const int blk = blockIdx.x;
    const long tok0 = (long)blk * TOK_PER_BLK;
    const float* xb = x + tok0 * CDIM;

    // Preload W into padded LDS (rows hit distinct banks; read broadcast later).
    for (int e = 0; e < NEXP; ++e)
        for (int k = tid; k < CDIM; k += 128)
            lw[e * W_STRIDE + k] = W[e * CDIM + k];

#if USE_WMMA
    const int lane = tid & 31;               // wave32
    const int wave = tid >> 5;
    const int m  = lane & 15;                // A row / D column index
    const int kh = lane >> 4;                // K-half selector (ISA layout)
    const int n  = m;                        // expert column; only n<3 live
    const bool live = (n < NEXP);
    const int  nsel = live ? n : 0;          // safe LDS row for dead lanes
    v8f acc = {0.f, 0.f, 0.f, 0.f, 0.f, 0.f, 0.f, 0.f};
#else
    float l0 = 0.f, l1 = 0.f, l2 = 0.f;
#endif

#if GFX_ASYNC
    // Prologue: kick off chunk 0 into buffer 0.
    {
        float* dst = xs;
        #pragma unroll
        for (int it = 0; it < 8; ++it) {
            const int s = it * 128 + tid;
            const int r = s >> 4;
            const int c = (s & 15) << 2;
            async_copy_b128(xb + (long)r * CDIM + c, dst + r * XS_STRIDE + c);
        }
    }
#endif

    for (int ci = 0; ci < NCHUNK; ++ci) {
        const int kc = ci * KCHUNK;
        float* xbuf = xs + (ci & 1) * XS_SIZE;

#if GFX_ASYNC
        if (ci + 1 < NCHUNK) {
            // Issue next chunk into the other buffer, then wait for current.
            float* dst = xs + ((ci + 1) & 1) * XS_SIZE;
            const int kn = kc + KCHUNK;
            #pragma unroll
            for (int it = 0; it < 8; ++it) {
                const int s = it * 128 + tid;
                const int r = s >> 4;
                const int c = (s & 15) << 2;
                async_copy_b128(xb + (long)r * CDIM + kn + c,
                                dst + r * XS_STRIDE + c);
            }
            WAIT_ASYNC(8);   // oldest 8 (current chunk) complete, in order
        } else {
            WAIT_ASYNC(0);
        }
        __syncthreads();
#else
        // Register-staged copy: all 8 loads in flight before any LDS store.
        float4 stg[8];
        #pragma unroll
        for (int it = 0; it < 8; ++it) {
            const int s = it * 128 + tid;
            const int r = s >> 4;
            const int c = (s & 15) << 2;
            stg[it] = *(const float4*)(xb + (long)r * CDIM + kc + c);
        }
        #pragma unroll
        for (int it = 0; it < 8; ++it) {
            const int s = it * 128 + tid;
            const int r = s >> 4;
            const int c = (s & 15) << 2;
            *(float4*)(&xbuf[r * XS_STRIDE + c]) = stg[it];
        }
        __syncthreads();
#endif

#if USE_WMMA
        const float* arow = xbuf + (wave * 16 + m) * XS_STRIDE;
        const float* wch  = lw + nsel * W_STRIDE + kc;
        #pragma unroll
        for (int ks = 0; ks < KCHUNK; ks += 4) {
            const int k = ks + 2 * kh;
            const float2 av = *(const float2*)(arow + k);   // A: x[m][k..k+1]
            const float2 wv = *(const float2*)(wch + k);    // B: W^T[k..k+1][n]
            v2f a; a[0] = av.x; a[1] = av.y;
            v2f b; b[0] = live ? wv.x : 0.f; b[1] = live ? wv.y : 0.f;
            acc = __builtin_amdgcn_wmma_f32_16x16x4_f32(
                false, a, false, b, (short)0, acc, false, false);
        }
#else
        if (tid < 64) {
            for (int k = 0; k < KCHUNK; ++k) {
                const float xv = xbuf[tid * XS_STRIDE + k];
                l0 = fmaf(xv, lw[0 * W_STRIDE + kc + k], l0);
                l1 = fmaf(xv, lw[1 * W_STRIDE + kc + k], l1);
                l2 = fmaf(xv, lw[2 * W_STRIDE + kc + k], l2);
            }
        }
#endif
        __syncthreads();
    }

#if USE_WMMA
    // D layout: VGPR v -> row (v + 8*kh), column n = lane&15.
    if (live) {
        #pragma unroll
        for (int v = 0; v < 8; ++v) lg[wave * 16 + v + 8 * kh][n] = acc[v];
    }
#else
    if (tid < 64) { lg[tid][0] = l0; lg[tid][1] = l1; lg[tid][2] = l2; }
#endif
    __syncthreads();

    // Per-token epilogue: softmax, argmax (lowest-index ties), lse^2.
    float zc = 0.f, q0 = 0.f, q1 = 0.f, q2 = 0.f;
    int c0 = 0, c1 = 0, c2 = 0;
    if (tid < 64) {
        const float a0 = lg[tid][0], a1 = lg[tid][1], a2 = lg[tid][2];
        const float mx = fmaxf(a0, fmaxf(a1, a2));
        const float e0 = expf(a0 - mx), e1 = expf(a1 - mx), e2 = expf(a2 - mx);
        const float ssum = e0 + e1 + e2;
        const float inv = 1.f / ssum;
        q0 = e0 * inv; q1 = e1 * inv; q2 = e2 * inv;
        const float lse = mx + logf(ssum);
        zc = lse * lse;
        int id = 0; float best = a0;
        if (a1 > best) { best = a1; id = 1; }
        if (a2 > best) { id = 2; }
        const long t = tok0 + tid;
        probs[t * 3 + 0] = q0; probs[t * 3 + 1] = q1; probs[t * 3 + 2] = q2;
        eidx[t] = id;
        c0 = (id == 0); c1 = (id == 1); c2 = (id == 2);
    }
    if (tid < 64) {
        redf[0][tid] = zc; redf[1][tid] = q0; redf[2][tid] = q1; redf[3][tid] = q2;
        redi[0][tid] = c0; redi[1][tid] = c1; redi[2][tid] = c2;
    }
    __syncthreads();
    if (tid == 0) {  // fixed-order sequential sum -> deterministic
        float sz = 0.f, s0 = 0.f, s1 = 0.f, s2 = 0.f;
        int k0 = 0, k1 = 0, k2 = 0;
        for (int i = 0; i < 64; ++i) {
            sz += redf[0][i]; s0 += redf[1][i]; s1 += redf[2][i]; s2 += redf[3][i];
            k0 += redi[0][i]; k1 += redi[1][i]; k2 += redi[2][i];
        }
        pz[blk] = sz;
        pP[blk * 3 + 0] = s0; pP[blk * 3 + 1] = s1; pP[blk * 3 + 2] = s2;
        pc[blk * 3 + 0] = k0; pc[blk * 3 + 1] = k1; pc[blk * 3 + 2] = k2;
    }
}

// ---------------------------------------------------------------------------
// Kernel 2: fold 512 block partials into totals (deterministic tree).
// ---------------------------------------------------------------------------
__global__ void __launch_bounds__(128)
reduce_kernel(const float* __restrict__ pz, const float* __restrict__ pP,
              const int* __restrict__ pc,
              float* __restrict__ tz, float* __restrict__ tP, int* __restrict__ tc)
{
    __shared__ float fz[128], f0[128], f1[128], f2[128];
    __shared__ int i0[128], i1[128], i2[128];
    const int t = threadIdx.x;
    float z = 0.f, a = 0.f, b = 0.f, c = 0.f;
    int x0 = 0, x1 = 0, x2 = 0;
    for (int i = t; i < NBLK; i += 128) {
        z += pz[i]; a += pP[i * 3]; b += pP[i * 3 + 1]; c += pP[i * 3 + 2];
        x0 += pc[i * 3]; x1 += pc[i * 3 + 1]; x2 += pc[i * 3 + 2];
    }
    fz[t] = z; f0[t] = a; f1[t] = b; f2[t] = c; i0[t] = x0; i1[t] = x1; i2[t] = x2;
    __syncthreads();
    for (int s = 64; s > 0; s >>= 1) {
        if (t < s) {
            fz[t] += fz[t + s]; f0[t] += f0[t + s]; f1[t] += f1[t + s]; f2[t] += f2[t + s];
            i0[t] += i0[t + s]; i1[t] += i1[t + s]; i2[t] += i2[t + s];
        }
        __syncthreads();
    }
    if (t == 0) {
        tz[0] = fz[0]; tP[0] = f0[0]; tP[1] = f1[0]; tP[2] = f2[0];
        tc[0] = i0[0]; tc[1] = i1[0]; tc[2] = i2[0];
    }
}

// ---------------------------------------------------------------------------
// Kernel 3: per-expert capacity threshold via bisection (L2-resident data).
// ---------------------------------------------------------------------------
__global__ void __launch_bounds__(256)
thresh_kernel(const float* __restrict__ probs, const int* __restrict__ eidx,
              const int* __restrict__ tc,
              float* __restrict__ thr_out, int* __restrict__ kept_out)
{
    const int e = blockIdx.x;
    const int t = threadIdx.x;
    __shared__ int red[256];
    __shared__ float sh_lo, sh_hi;

    const int assigned = tc[e];   // uniform across block
    float thr;
    if (assigned <= CAP) {
        thr = -1.0f;              // under capacity: keep every assigned token
    } else {
        if (t == 0) { sh_lo = 0.f; sh_hi = 1.f; }
        __syncthreads();
        for (int it = 0; it < 40; ++it) {
            const float mid = 0.5f * (sh_lo + sh_hi);
            int cnt = 0;
            for (int i = t; i < NTOK; i += 256)
                if (eidx[i] == e && probs[i * 3 + e] > mid) ++cnt;
            red[t] = cnt;
            __syncthreads();
            for (int s = 128; s > 0; s >>= 1) {
                if (t < s) red[t] += red[t + s];
                __syncthreads();
            }
            if (t == 0) { if (red[0] > CAP) sh_lo = mid; else sh_hi = mid; }
            __syncthreads();
        }
        thr = sh_hi;              // count(> thr) == CAP for distinct values
    }

    int cnt = 0;                  // exact kept count for aux-loss f
    for (int i = t; i < NTOK; i += 256)
        if (eidx[i] == e && probs[i * 3 + e] > thr) ++cnt;
    red[t] = cnt;
    __syncthreads();
    for (int s = 128; s > 0; s >>= 1) {
        if (t < s) red[t] += red[t + s];
        __syncthreads();
    }
    if (t == 0) { thr_out[e] = thr; kept_out[e] = red[0]; }
}

// ---------------------------------------------------------------------------
// Kernel 4: gates + dispatch mask -> d_out
// ---------------------------------------------------------------------------
__global__ void __launch_bounds__(256)
finalize_kernel(const float* __restrict__ probs, const int* __restrict__ eidx,
                const float* __restrict__ thr,
                float* __restrict__ out_gates, float* __restrict__ out_mask)
{
    const int t = blockIdx.x * 256 + threadIdx.x;
    if (t >= NTOK) return;
    const float p0 = probs[t * 3 + 0], p1 = probs[t * 3 + 1], p2 = probs[t * 3 + 2];
    const int id = eidx[t];
    const float d0 = (id == 0 && p0 > thr[0]) ? 1.f : 0.f;
    const float d1 = (id == 1 && p1 > thr[1]) ? 1.f : 0.f;
    const float d2 = (id == 2 && p2 > thr[2]) ? 1.f : 0.f;
    const float g0 = p0 * d0, g1 = p1 * d1, g2 = p2 * d2;
    const float inv = 1.f / (g0 + g1 + g2 + 1e-6f);
    out_gates[t * 3 + 0] = g0 * inv;
    out_gates[t * 3 + 1] = g1 * inv;
    out_gates[t * 3 + 2] = g2 * inv;
    out_mask[t * 3 + 0] = d0;
    out_mask[t * 3 + 1] = d1;
    out_mask[t * 3 + 2] = d2;
}

// ---------------------------------------------------------------------------
// Kernel 5: scalar loss = aux + z
// ---------------------------------------------------------------------------
__global__ void loss_kernel(const float* __restrict__ tz, const float* __restrict__ tP,
                            const int* __restrict__ kept, float* __restrict__ out_loss)
{
    if (blockIdx.x == 0 && threadIdx.x == 0) {
        const float invN = 1.0f / (float)NTOK;
        float aux = 0.f;
        for (int e = 0; e < NEXP; ++e)
            aux += ((float)kept[e] * invN) * (tP[e] * invN);
        aux *= 0.1f * (float)NEXP;               // AUX_W * E
        const float z = 0.001f * tz[0] * invN;   // Z_W * mean(lse^2)
        out_loss[0] = aux + z;
    }
}

extern "C" void kernel_launch(void* const* d_in, const int* in_sizes, int n_in,
                              void* d_out, int out_size, void* d_ws, size_t ws_size,
                              hipStream_t stream)
{
    (void)in_sizes; (void)n_in; (void)out_size; (void)ws_size;
    const float* x = (const float*)d_in[0];
    const float* W = (const float*)d_in[1];
    float* out = (float*)d_out;

    // workspace carve-up (~530 KB)
    float* ws_probs = (float*)d_ws;                // NTOK*3 floats
    int*   ws_idx   = (int*)(ws_probs + NTOK * 3); // NTOK ints
    float* ws_pz    = (float*)(ws_idx + NTOK);     // NBLK
    float* ws_pP    = ws_pz + NBLK;                // NBLK*3
    int*   ws_pc    = (int*)(ws_pP + NBLK * 3);    // NBLK*3
    float* ws_tz    = (float*)(ws_pc + NBLK * 3);  // 1
    float* ws_tP    = ws_tz + 1;                   // 3
    int*   ws_tc    = (int*)(ws_tP + 3);           // 3
    float* ws_thr   = (float*)(ws_tc + 3);         // 3
    int*   ws_kept  = (int*)(ws_thr + 3);          // 3

    router_kernel<<<NBLK, 128, 0, stream>>>(x, W, ws_probs, ws_idx, ws_pz, ws_pP, ws_pc);
    reduce_kernel<<<1, 128, 0, stream>>>(ws_pz, ws_pP, ws_pc, ws_tz, ws_tP, ws_tc);
    thresh_kernel<<<NEXP, 256, 0, stream>>>(ws_probs, ws_idx, ws_tc, ws_thr, ws_kept);
    finalize_kernel<<<NTOK / 256, 256, 0, stream>>>(ws_probs, ws_idx, ws_thr,
                                                    out, out + NTOK * 3);
    loss_kernel<<<1, 32, 0, stream>>>(ws_tz, ws_tP, ws_kept, out + NTOK * 6);
}